// PGCN_3564822855941
// MI455X (gfx1250) — compile-verified
//
#include <hip/hip_runtime.h>
#include <math.h>

typedef float v2f __attribute__((ext_vector_type(2)));
typedef float v8f __attribute__((ext_vector_type(8)));

// ---------------------------------------------------------------- utilities

__global__ void fill_kernel(float* __restrict__ p, float v, int n) {
  for (int i = blockIdx.x * blockDim.x + threadIdx.x; i < n;
       i += gridDim.x * blockDim.x)
    p[i] = v;
}

__global__ void count_kernel(const int* __restrict__ idx, int E,
                             float* __restrict__ deg) {
  for (int i = blockIdx.x * blockDim.x + threadIdx.x; i < E;
       i += gridDim.x * blockDim.x)
    atomicAdd(&deg[idx[i]], 1.0f);
}

__global__ void dinv_kernel(float* __restrict__ d, int n) {
  for (int i = blockIdx.x * blockDim.x + threadIdx.x; i < n;
       i += gridDim.x * blockDim.x) {
    float x = d[i];
    d[i] = (x > 0.0f) ? (1.0f / sqrtf(x)) : 0.0f;
  }
}

// ------------------------------------------------- fp32 WMMA GEMM  C = A*B
// One wave computes one 16x16 tile of C, accumulating with
// V_WMMA_F32_16X16X4_F32 over K in steps of 4.  M,N multiples of 16, K of 4.
// A: [M,K] row-major, B: [K,N] row-major, C: [M,N] row-major.
__global__ __launch_bounds__(128) void gemm_wmma_f32(
    const float* __restrict__ A, const float* __restrict__ B,
    float* __restrict__ C, int M, int N, int K) {
  const int wave = threadIdx.x >> 5;
  const int lane = threadIdx.x & 31;
  const int ntN = N >> 4;
  const int ntiles = (M >> 4) * ntN;
  const int tile = blockIdx.x * (blockDim.x >> 5) + wave;
  if (tile >= ntiles) return;  // whole-wave exit: EXEC stays all-ones
  const int mt = tile / ntN;
  const int nt = tile - mt * ntN;
  const int hi = lane >> 4;   // 0..1  (K sub-pair select)
  const int lo = lane & 15;   // m for A, n for B/C/D

  const float* arow = A + (size_t)(mt * 16 + lo) * K;        // A[m][*]
  const float* bcol = B + (size_t)(nt * 16 + lo);            // B[*][n]
  v8f acc = {};

  for (int kb = 0; kb < K; kb += 4) {
    const int k0 = kb + 2 * hi;
    v2f a = { arow[k0], arow[k0 + 1] };                      // A[m][k0..k0+1]
    v2f b = { bcol[(size_t)k0 * N], bcol[(size_t)(k0 + 1) * N] };
    acc = __builtin_amdgcn_wmma_f32_16x16x4_f32(
        /*neg_a=*/false, a, /*neg_b=*/false, b,
        /*c_mod=*/(short)0, acc, /*reuse_a=*/false, /*reuse_b=*/false);
  }

  // D VGPR v, lane hi*16+n  ->  D[v + 8*hi][n]
  float* crow = C + (size_t)(mt * 16 + 8 * hi) * N + nt * 16 + lo;
#pragma unroll
  for (int v = 0; v < 8; ++v) crow[(size_t)v * N] = acc[v];
}

// -------------------------------------------- edge gather/scale/scatter-add
// thread i handles 4 consecutive features of one edge message:
//   acc[col[e]][c..c+3] += h[row[e]][c..c+3] * dinv_r[row[e]] * dinv_c[col[e]]
__global__ void scatter_kernel(const int* __restrict__ rows,
                               const int* __restrict__ cols, int E,
                               const float* __restrict__ dinv_r,
                               const float* __restrict__ dinv_c,
                               const float* __restrict__ h,
                               float* __restrict__ acc, int F) {
  const int chunks = F >> 2;
  const int total = E * chunks;
  for (int i = blockIdx.x * blockDim.x + threadIdx.x; i < total;
       i += gridDim.x * blockDim.x) {
    const int e = i / chunks;
    const int c = (i - e * chunks) << 2;
    const int r = rows[e];
    const int d = cols[e];
    const float nrm = dinv_r[r] * dinv_c[d];
    const float4 hv = *(const float4*)(h + (size_t)r * F + c);
    float* o = acc + (size_t)d * F + c;
    atomicAdd(o + 0, hv.x * nrm);
    atomicAdd(o + 1, hv.y * nrm);
    atomicAdd(o + 2, hv.z * nrm);
    atomicAdd(o + 3, hv.w * nrm);
  }
}

// ---------------- combine: acc + self-loop term + both relation biases, ReLU
// hself may be null-equivalent by passing dinv of all-zero — here homo always
// contributes, so it is always valid.
__global__ void combine_relu_kernel(const float* __restrict__ acc,
                                    const float* __restrict__ hself,
                                    const float* __restrict__ dinv,
                                    const float* __restrict__ ba,
                                    const float* __restrict__ bb,
                                    float* __restrict__ out, int Nn, int F) {
  const int total = Nn * F;
  for (int i = blockIdx.x * blockDim.x + threadIdx.x; i < total;
       i += gridDim.x * blockDim.x) {
    const int n = i / F;
    const int f = i - n * F;
    const float di = dinv[n];
    float v = acc[i] + hself[i] * (di * di) + ba[f] + bb[f];
    out[i] = v > 0.0f ? v : 0.0f;
  }
}

// ---------------------------------------------------------------- host side

static inline unsigned nblk(long long n, int tpb) {
  long long b = (n + tpb - 1) / tpb;
  if (b < 1) b = 1;
  if (b > 1048576) b = 1048576;
  return (unsigned)b;
}

extern "C" void kernel_launch(void* const* d_in, const int* in_sizes, int n_in,
                              void* d_out, int out_size, void* d_ws,
                              size_t ws_size, hipStream_t stream) {
  const float* xg = (const float*)d_in[0];
  const float* xd = (const float*)d_in[1];
  const int* ei_gg = (const int*)d_in[2];
  const int* ei_dd = (const int*)d_in[3];
  const int* ei_gd = (const int*)d_in[4];
  const int* ei_dg = (const int*)d_in[5];
  const float* W1_gg = (const float*)d_in[6];
  const float* b1_gg = (const float*)d_in[7];
  const float* W2_gg = (const float*)d_in[8];
  const float* b2_gg = (const float*)d_in[9];
  const float* W1_dd = (const float*)d_in[10];
  const float* b1_dd = (const float*)d_in[11];
  const float* W2_dd = (const float*)d_in[12];
  const float* b2_dd = (const float*)d_in[13];
  const float* W1_gd = (const float*)d_in[14];
  const float* b1_gd = (const float*)d_in[15];
  const float* W2_gd = (const float*)d_in[16];
  const float* b2_gd = (const float*)d_in[17];
  const float* W1_dg = (const float*)d_in[18];
  const float* b1_dg = (const float*)d_in[19];
  const float* W2_dg = (const float*)d_in[20];
  const float* b2_dg = (const float*)d_in[21];

  const int NG = 50000, ND = 10000;
  const int E_gg = in_sizes[2] / 2;  // 1600000
  const int E_dd = in_sizes[3] / 2;  // 320000
  const int E_gd = in_sizes[4] / 2;  // 500000
  const int E_dg = in_sizes[5] / 2;  // 500000
  const int DIN = 64, HID = 64, OUT = 32;
  const int TPB = 256;

  // ---- workspace carve-out (floats) ----
  float* ws = (float*)d_ws;
  size_t off = 0;
  auto take = [&](size_t n) { float* p = ws + off; off += n; return p; };
  float* dinv_gg = take(NG);          // homo gene degrees^-1/2 (incl self loop)
  float* dinv_dd = take(ND);
  float* dinv_gd_s = take(NG);        // gd: source=gene, dest=disease
  float* dinv_gd_d = take(ND);
  float* dinv_dg_s = take(ND);        // dg: source=disease, dest=gene
  float* dinv_dg_d = take(NG);
  float* h_gg = take((size_t)NG * HID);
  float* h_gd = take((size_t)NG * HID);
  float* h_dd = take((size_t)ND * HID);
  float* h_dg = take((size_t)ND * HID);
  float* acc_g = take((size_t)NG * HID);
  float* acc_d = take((size_t)ND * HID);
  float* hg = take((size_t)NG * HID);
  float* hd = take((size_t)ND * HID);
  // layer-2 projections overlay freed layer-1 h buffers
  float* h2_gg = h_gg;                      // NG*OUT
  float* h2_gd = h_gg + (size_t)NG * OUT;   // NG*OUT  (fits in h_gg's NG*64)
  float* h2_dd = h_gd;                      // ND*OUT
  float* h2_dg = h_gd + (size_t)ND * OUT;   // ND*OUT

  // ---- 1) degree / normalization (edge-structure only, shared by layers) --
  // homo: deg seeded with 1.0 for the self loop, counted over col
  fill_kernel<<<nblk(NG, TPB), TPB, 0, stream>>>(dinv_gg, 1.0f, NG);
  count_kernel<<<nblk(E_gg, TPB), TPB, 0, stream>>>(ei_gg + E_gg, E_gg, dinv_gg);
  dinv_kernel<<<nblk(NG, TPB), TPB, 0, stream>>>(dinv_gg, NG);

  fill_kernel<<<nblk(ND, TPB), TPB, 0, stream>>>(dinv_dd, 1.0f, ND);
  count_kernel<<<nblk(E_dd, TPB), TPB, 0, stream>>>(ei_dd + E_dd, E_dd, dinv_dd);
  dinv_kernel<<<nblk(ND, TPB), TPB, 0, stream>>>(dinv_dd, ND);

  // bipartite gd
  fill_kernel<<<nblk(NG, TPB), TPB, 0, stream>>>(dinv_gd_s, 0.0f, NG);
  count_kernel<<<nblk(E_gd, TPB), TPB, 0, stream>>>(ei_gd, E_gd, dinv_gd_s);
  dinv_kernel<<<nblk(NG, TPB), TPB, 0, stream>>>(dinv_gd_s, NG);
  fill_kernel<<<nblk(ND, TPB), TPB, 0, stream>>>(dinv_gd_d, 0.0f, ND);
  count_kernel<<<nblk(E_gd, TPB), TPB, 0, stream>>>(ei_gd + E_gd, E_gd, dinv_gd_d);
  dinv_kernel<<<nblk(ND, TPB), TPB, 0, stream>>>(dinv_gd_d, ND);

  // bipartite dg
  fill_kernel<<<nblk(ND, TPB), TPB, 0, stream>>>(dinv_dg_s, 0.0f, ND);
  count_kernel<<<nblk(E_dg, TPB), TPB, 0, stream>>>(ei_dg, E_dg, dinv_dg_s);
  dinv_kernel<<<nblk(ND, TPB), TPB, 0, stream>>>(dinv_dg_s, ND);
  fill_kernel<<<nblk(NG, TPB), TPB, 0, stream>>>(dinv_dg_d, 0.0f, NG);
  count_kernel<<<nblk(E_dg, TPB), TPB, 0, stream>>>(ei_dg + E_dg, E_dg, dinv_dg_d);
  dinv_kernel<<<nblk(NG, TPB), TPB, 0, stream>>>(dinv_dg_d, NG);

  // ---- 2) layer-1 projections (WMMA fp32) ----
  {
    int t1 = (NG / 16) * (HID / 16), t2 = (ND / 16) * (HID / 16);
    gemm_wmma_f32<<<(t1 + 3) / 4, 128, 0, stream>>>(xg, W1_gg, h_gg, NG, HID, DIN);
    gemm_wmma_f32<<<(t1 + 3) / 4, 128, 0, stream>>>(xg, W1_gd, h_gd, NG, HID, DIN);
    gemm_wmma_f32<<<(t2 + 3) / 4, 128, 0, stream>>>(xd, W1_dd, h_dd, ND, HID, DIN);
    gemm_wmma_f32<<<(t2 + 3) / 4, 128, 0, stream>>>(xd, W1_dg, h_dg, ND, HID, DIN);
  }

  // ---- 3) layer-1 aggregation ----
  fill_kernel<<<nblk((size_t)NG * HID, TPB), TPB, 0, stream>>>(acc_g, 0.0f, NG * HID);
  fill_kernel<<<nblk((size_t)ND * HID, TPB), TPB, 0, stream>>>(acc_d, 0.0f, ND * HID);

  scatter_kernel<<<nblk((long long)E_gg * (HID / 4), TPB), TPB, 0, stream>>>(
      ei_gg, ei_gg + E_gg, E_gg, dinv_gg, dinv_gg, h_gg, acc_g, HID);
  scatter_kernel<<<nblk((long long)E_dg * (HID / 4), TPB), TPB, 0, stream>>>(
      ei_dg, ei_dg + E_dg, E_dg, dinv_dg_s, dinv_dg_d, h_dg, acc_g, HID);
  scatter_kernel<<<nblk((long long)E_dd * (HID / 4), TPB), TPB, 0, stream>>>(
      ei_dd, ei_dd + E_dd, E_dd, dinv_dd, dinv_dd, h_dd, acc_d, HID);
  scatter_kernel<<<nblk((long long)E_gd * (HID / 4), TPB), TPB, 0, stream>>>(
      ei_gd, ei_gd + E_gd, E_gd, dinv_gd_s, dinv_gd_d, h_gd, acc_d, HID);

  // ---- 4) layer-1 combine: self-loop + biases + ReLU ----
  combine_relu_kernel<<<nblk((size_t)NG * HID, TPB), TPB, 0, stream>>>(
      acc_g, h_gg, dinv_gg, b1_gg, b1_dg, hg, NG, HID);
  combine_relu_kernel<<<nblk((size_t)ND * HID, TPB), TPB, 0, stream>>>(
      acc_d, h_dd, dinv_dd, b1_dd, b1_gd, hd, ND, HID);

  // ---- 5) layer-2 projections (WMMA fp32) ----
  {
    int t1 = (NG / 16) * (OUT / 16), t2 = (ND / 16) * (OUT / 16);
    gemm_wmma_f32<<<(t1 + 3) / 4, 128, 0, stream>>>(hg, W2_gg, h2_gg, NG, OUT, HID);
    gemm_wmma_f32<<<(t1 + 3) / 4, 128, 0, stream>>>(hg, W2_gd, h2_gd, NG, OUT, HID);
    gemm_wmma_f32<<<(t2 + 3) / 4, 128, 0, stream>>>(hd, W2_dd, h2_dd, ND, OUT, HID);
    gemm_wmma_f32<<<(t2 + 3) / 4, 128, 0, stream>>>(hd, W2_dg, h2_dg, ND, OUT, HID);
  }

  // ---- 6) layer-2 aggregation directly into d_out ----
  float* og = (float*)d_out;
  float* od = og + (size_t)NG * OUT;
  fill_kernel<<<nblk((size_t)(NG + ND) * OUT, TPB), TPB, 0, stream>>>(
      og, 0.0f, (NG + ND) * OUT);

  scatter_kernel<<<nblk((long long)E_gg * (OUT / 4), TPB), TPB, 0, stream>>>(
      ei_gg, ei_gg + E_gg, E_gg, dinv_gg, dinv_gg, h2_gg, og, OUT);
  scatter_kernel<<<nblk((long long)E_dg * (OUT / 4), TPB), TPB, 0, stream>>>(
      ei_dg, ei_dg + E_dg, E_dg, dinv_dg_s, dinv_dg_d, h2_dg, og, OUT);
  scatter_kernel<<<nblk((long long)E_dd * (OUT / 4), TPB), TPB, 0, stream>>>(
      ei_dd, ei_dd + E_dd, E_dd, dinv_dd, dinv_dd, h2_dd, od, OUT);
  scatter_kernel<<<nblk((long long)E_gd * (OUT / 4), TPB), TPB, 0, stream>>>(
      ei_gd, ei_gd + E_gd, E_gd, dinv_gd_s, dinv_gd_d, h2_gd, od, OUT);

  // ---- 7) layer-2 combine (in place) + final ReLU ----
  combine_relu_kernel<<<nblk((size_t)NG * OUT, TPB), TPB, 0, stream>>>(
      og, h2_gg, dinv_gg, b2_gg, b2_dg, og, NG, OUT);
  combine_relu_kernel<<<nblk((size_t)ND * OUT, TPB), TPB, 0, stream>>>(
      od, h2_dd, dinv_dd, b2_dd, b2_gd, od, ND, OUT);
}